// Decoder_64012192580153
// MI455X (gfx1250) — compile-verified
//
#include <hip/hip_runtime.h>
#include <hip/hip_bf16.h>

typedef __attribute__((ext_vector_type(2))) float v2f;
typedef __attribute__((ext_vector_type(8))) float v8f;

#define NN 2048
#define DD 32
#define HH 64

// ---------------------------------------------------------------------------
// Kernel 1: A' = Z @ W1[:32] + b1,  B = Z @ W1[32:]  via V_WMMA_F32_16X16X4_F32
// One wave computes one 16x16 f32 tile with 8 WMMA steps (K=32 in chunks of 4).
// ISA layouts (CDNA5 ISA):
//   A 16x4 f32 : lane&15 = M; VGPR0 = K={0|2}, VGPR1 = K={1|3} (lane>=16 -> +2)
//   B 4x16 f32 : lane&15 = N; same K split across lane halves
//   C/D 16x16  : VGPR r -> M = r (lanes 0-15) / r+8 (lanes 16-31), N = lane&15
// ---------------------------------------------------------------------------
__global__ __launch_bounds__(256) void precompute_wmma(
    const float* __restrict__ Z, const float* __restrict__ W1,
    const float* __restrict__ b1, float* __restrict__ Apb,
    float* __restrict__ Bm)
{
    const int wave    = threadIdx.x >> 5;
    const int lane    = threadIdx.x & 31;
    const int tileIdx = blockIdx.x * 8 + wave;      // 0..1023
    const bool isB    = tileIdx >= 512;             // second half -> B matrix
    const int tt      = tileIdx & 511;
    const int rowBase = (tt >> 2) * 16;             // 128 row tiles
    const int colBase = (tt & 3) * 16;              // 4 col tiles (H=64)
    const int kbase   = isB ? DD : 0;               // W1 row offset
    const int c0      = lane & 15;                  // M for A-frag, N for B-frag
    const int hi      = lane >> 4;                  // K half select

    v8f acc = {};
#pragma unroll
    for (int kk = 0; kk < 8; ++kk) {
        const int k = kk * 4 + hi * 2;
        v2f a, b;
        a.x = Z[(rowBase + c0) * DD + k];
        a.y = Z[(rowBase + c0) * DD + k + 1];
        b.x = W1[(kbase + k) * HH + colBase + c0];
        b.y = W1[(kbase + k + 1) * HH + colBase + c0];
        // 8 args: (neg_a, A, neg_b, B, c_mod, C, reuse_a, reuse_b)
        acc = __builtin_amdgcn_wmma_f32_16x16x4_f32(
            false, a, false, b, (short)0, acc, false, false);
    }

    const float bias = isB ? 0.0f : b1[colBase + c0];
    float* dst = isB ? Bm : Apb;
#pragma unroll
    for (int r = 0; r < 8; ++r) {
        dst[(rowBase + r + hi * 8) * HH + colBase + c0] = acc[r] + bias;
    }
}

// ---------------------------------------------------------------------------
// Kernel 2: out[i,j] = sigmoid( sum_h relu(A'[i,h] + B[j,h]) * W2[h] + b2 )
// 64x64 output tile per 256-thread block (8 wave32), 4x4 micro-tile per
// thread. Panels staged in LDS transposed ([h][row]) so the inner loop is
// two ds_load_b128 per h against 48 f32 VALU ops (add/max/fma).
// ---------------------------------------------------------------------------
__global__ __launch_bounds__(256) void pair_sigmoid(
    const float* __restrict__ Apb, const float* __restrict__ Bm,
    const float* __restrict__ W2, const float* __restrict__ b2,
    float* __restrict__ out)
{
    __shared__ float As[HH][64];   // [h][i-in-tile]
    __shared__ float Bs[HH][64];   // [h][j-in-tile]
    __shared__ float w2s[HH];

    const int tid = threadIdx.x;
    const int i0  = blockIdx.y * 64;
    const int j0  = blockIdx.x * 64;

    // Stage panels (transposed) : 4096 elems each, 16 per thread.
    for (int e = tid; e < 64 * HH; e += 256) {
        const int r = e >> 6;      // row within tile
        const int h = e & 63;
        As[h][r] = Apb[(size_t)(i0 + r) * HH + h];
        Bs[h][r] = Bm[(size_t)(j0 + r) * HH + h];
    }
    if (tid < HH) w2s[tid] = W2[tid];
    __syncthreads();

    const int ti = (tid >> 4) * 4;     // 16x16 thread grid, 4x4 outputs each
    const int tj = (tid & 15) * 4;
    const float bb = b2[0];

    float acc[4][4];
#pragma unroll
    for (int ii = 0; ii < 4; ++ii)
#pragma unroll
        for (int jj = 0; jj < 4; ++jj) acc[ii][jj] = bb;

#pragma unroll 4
    for (int h = 0; h < HH; ++h) {
        const float4 av = *(const float4*)&As[h][ti];
        const float4 bv = *(const float4*)&Bs[h][tj];
        const float w = w2s[h];
        const float a_[4] = {av.x, av.y, av.z, av.w};
        const float b_[4] = {bv.x, bv.y, bv.z, bv.w};
#pragma unroll
        for (int ii = 0; ii < 4; ++ii)
#pragma unroll
            for (int jj = 0; jj < 4; ++jj) {
                const float t = fmaxf(a_[ii] + b_[jj], 0.0f);
                acc[ii][jj] = fmaf(t, w, acc[ii][jj]);
            }
    }

#pragma unroll
    for (int ii = 0; ii < 4; ++ii) {
        float4 o;
        o.x = 1.0f / (1.0f + __expf(-acc[ii][0]));
        o.y = 1.0f / (1.0f + __expf(-acc[ii][1]));
        o.z = 1.0f / (1.0f + __expf(-acc[ii][2]));
        o.w = 1.0f / (1.0f + __expf(-acc[ii][3]));
        *(float4*)&out[(size_t)(i0 + ti + ii) * NN + (j0 + tj)] = o;
    }
}

extern "C" void kernel_launch(void* const* d_in, const int* in_sizes, int n_in,
                              void* d_out, int out_size, void* d_ws, size_t ws_size,
                              hipStream_t stream) {
    (void)in_sizes; (void)n_in; (void)out_size; (void)ws_size;
    const float* Z  = (const float*)d_in[0];   // [2048, 32]
    const float* W1 = (const float*)d_in[1];   // [64, 64]
    const float* b1 = (const float*)d_in[2];   // [64]
    const float* W2 = (const float*)d_in[3];   // [64, 1]
    const float* b2 = (const float*)d_in[4];   // [1]
    float* out = (float*)d_out;                // [2048, 2048]

    float* Apb = (float*)d_ws;                 // [2048, 64]  (A + b1)
    float* Bm  = Apb + (size_t)NN * HH;        // [2048, 64]

    // 1024 16x16 tiles total (A' and B), 8 waves per block -> 128 blocks.
    precompute_wmma<<<128, 256, 0, stream>>>(Z, W1, b1, Apb, Bm);

    dim3 grid(NN / 64, NN / 64);               // 32 x 32 output tiles
    pair_sigmoid<<<grid, 256, 0, stream>>>(Apb, Bm, W2, b2, out);
}